// GIN_78898549227759
// MI455X (gfx1250) — compile-verified
//
#include <hip/hip_runtime.h>
#include <hip/hip_bf16.h>
#include <math.h>

typedef __attribute__((ext_vector_type(2))) float v2f;
typedef __attribute__((ext_vector_type(4))) float v4f;
typedef __attribute__((ext_vector_type(8))) float v8f;

// ---------------------------------------------------------------- utilities

__global__ void zero_kernel(float* __restrict__ p, long long n) {
    long long i = (long long)blockIdx.x * blockDim.x + threadIdx.x;
    if (i < n) p[i] = 0.0f;
}

// GIN aggregation: agg[dst] += x[src]  (one wave per edge, 2 feats/lane)
// also accumulates in-degree for GCN normalization.
__global__ void gin_edge_kernel(const float* __restrict__ x,
                                const int* __restrict__ src,
                                const int* __restrict__ dst,
                                float* __restrict__ agg,
                                float* __restrict__ deg,
                                int E) {
    long long idx = (long long)blockIdx.x * blockDim.x + threadIdx.x;
    int e = (int)(idx >> 5);
    int f = (int)(idx & 31);
    if (e >= E) return;
    int s = src[e], d = dst[e];
    atomicAdd(&agg[(long long)d * 64 + f],      x[(long long)s * 64 + f]);
    atomicAdd(&agg[(long long)d * 64 + f + 32], x[(long long)s * 64 + f + 32]);
    if (f == 0) atomicAdd(&deg[d], 1.0f);
}

// deg -> rsqrt(deg + 1)   (self loop adds 1)
__global__ void dinv_kernel(float* __restrict__ deg, int n) {
    int i = blockIdx.x * blockDim.x + threadIdx.x;
    if (i < n) deg[i] = rsqrtf(deg[i] + 1.0f);
}

// GCN: out[i,f] = hw[i,f]*dinv[i]^2 + b[f]   (self-loop term + bias)
__global__ void gcn_init_kernel(const float* __restrict__ hw,
                                const float* __restrict__ dinv,
                                const float* __restrict__ b,
                                float* __restrict__ out, long long n64) {
    long long i = (long long)blockIdx.x * blockDim.x + threadIdx.x;
    if (i >= n64) return;
    int row = (int)(i >> 6);
    float di = dinv[row];
    out[i] = hw[i] * di * di + b[i & 63];
}

// GCN scatter: out[dst] += hw[src] * dinv[src]*dinv[dst]
__global__ void gcn_edge_kernel(const float* __restrict__ hw,
                                const int* __restrict__ src,
                                const int* __restrict__ dst,
                                const float* __restrict__ dinv,
                                float* __restrict__ out,
                                int E) {
    long long idx = (long long)blockIdx.x * blockDim.x + threadIdx.x;
    int e = (int)(idx >> 5);
    int f = (int)(idx & 31);
    if (e >= E) return;
    int s = src[e], d = dst[e];
    float nrm = dinv[s] * dinv[d];
    atomicAdd(&out[(long long)d * 64 + f],      hw[(long long)s * 64 + f]      * nrm);
    atomicAdd(&out[(long long)d * 64 + f + 32], hw[(long long)s * 64 + f + 32] * nrm);
}

// ---------------------------------------------------------------- WMMA GEMM
// out = act2( act1( (in [+in2]) @ W1 + b1 ) @ W2 + b2 )     (W2 optional)
// One wave computes a 16-row tile; K=64 via 16x V_WMMA_F32_16X16X4_F32.
// A-frag (16x4 f32): lanes 0-15 hold M=0..15 with K={k,k+1}; lanes 16-31 K={k+2,k+3}.
// B-frag (4x16 f32): lanes 0-15 hold N=0..15 with K={k,k+1}; lanes 16-31 K={k+2,k+3}.
// D (16x16 f32): VGPR j -> row j (lanes<16) / j+8 (lanes>=16), col = lane&15.
__launch_bounds__(256)
__global__ void gemm64_wmma_kernel(const float* __restrict__ in,
                                   const float* __restrict__ in2,
                                   const float* __restrict__ W1,
                                   const float* __restrict__ b1,
                                   const float* __restrict__ W2,
                                   const float* __restrict__ b2,
                                   float* __restrict__ out,
                                   int n, int relu1, int relu2) {
    __shared__ float w1s[64 * 64];
    __shared__ float w2s[64 * 64];
    __shared__ float b1s[64];
    __shared__ float b2s[64];
    __shared__ float stage[8][16 * 68];   // per-wave 16x64 restage, padded stride

    int t = threadIdx.x;
    for (int i = t; i < 64 * 64 / 4; i += 256) {
        ((v4f*)w1s)[i] = ((const v4f*)W1)[i];
        if (W2) ((v4f*)w2s)[i] = ((const v4f*)W2)[i];
    }
    if (t < 64) {
        b1s[t] = b1 ? b1[t] : 0.0f;
        b2s[t] = b2 ? b2[t] : 0.0f;
    }
    __syncthreads();

    int wave = t >> 5, lane = t & 31;
    int ntiles = (n + 15) >> 4;
    int tile = blockIdx.x * 8 + wave;
    if (tile >= ntiles) tile = ntiles - 1;   // clamp: duplicate identical work
    int row0 = tile * 16;

    int am = lane & 15;            // M index
    int ak = (lane >> 4) * 2;      // K sub-offset within a K=4 chunk
    int bn = lane & 15;            // N index
    int bk = (lane >> 4) * 2;

    // A fragments for all 16 K-chunks (32 VGPRs)
    v2f a[16];
    const float* ain  = in + (long long)(row0 + am) * 64;
    const float* ain2 = in2 ? in2 + (long long)(row0 + am) * 64 : nullptr;
#pragma unroll
    for (int k = 0; k < 16; ++k) {
        v2f v = *(const v2f*)(ain + k * 4 + ak);
        if (in2) {
            v2f u = *(const v2f*)(ain2 + k * 4 + ak);
            v.x += u.x; v.y += u.y;
        }
        a[k] = v;
    }

    float* st = &stage[wave][0];
    int rb = (lane < 16) ? 0 : 8;

    // ---- layer 1
#pragma unroll
    for (int nt = 0; nt < 4; ++nt) {
        v8f c = {};
#pragma unroll
        for (int k = 0; k < 16; ++k) {
            int kr = k * 4 + bk;
            v2f b;
            b.x = w1s[kr * 64 + nt * 16 + bn];
            b.y = w1s[(kr + 1) * 64 + nt * 16 + bn];
            c = __builtin_amdgcn_wmma_f32_16x16x4_f32(false, a[k], false, b,
                                                      (short)0, c, false, false);
        }
        float bias = b1s[nt * 16 + bn];
        int colw = nt * 16 + bn;
#pragma unroll
        for (int j = 0; j < 8; ++j) {
            float v = c[j] + bias;
            if (relu1) v = fmaxf(v, 0.0f);
            if (W2) st[(rb + j) * 68 + colw] = v;                       // restage
            else    out[(long long)(row0 + rb + j) * 64 + colw] = v;    // store
        }
    }

    // ---- layer 2 (optional)
    if (W2) {
        __syncthreads();
        v2f a2[16];
#pragma unroll
        for (int k = 0; k < 16; ++k) {
            a2[k].x = st[am * 68 + k * 4 + ak];
            a2[k].y = st[am * 68 + k * 4 + ak + 1];
        }
#pragma unroll
        for (int nt = 0; nt < 4; ++nt) {
            v8f c = {};
#pragma unroll
            for (int k = 0; k < 16; ++k) {
                int kr = k * 4 + bk;
                v2f b;
                b.x = w2s[kr * 64 + nt * 16 + bn];
                b.y = w2s[(kr + 1) * 64 + nt * 16 + bn];
                c = __builtin_amdgcn_wmma_f32_16x16x4_f32(false, a2[k], false, b,
                                                          (short)0, c, false, false);
            }
            float bias = b2s[nt * 16 + bn];
            int colw = nt * 16 + bn;
#pragma unroll
            for (int j = 0; j < 8; ++j) {
                float v = c[j] + bias;
                if (relu2) v = fmaxf(v, 0.0f);
                out[(long long)(row0 + rb + j) * 64 + colw] = v;
            }
        }
    }
}

// ---------------------------------------------------------------- final head
// logits = xemb @ Wl(64x40) + bl ; out = log_softmax(logits). Wl zero-padded
// to 64x48 in LDS so 3 WMMA N-tiles cover it; softmax via per-wave LDS stage.
__launch_bounds__(256)
__global__ void final_kernel(const float* __restrict__ xemb,
                             const float* __restrict__ Wl,
                             const float* __restrict__ bl,
                             float* __restrict__ out, int n) {
    __shared__ float wls[64 * 48];
    __shared__ float bs[48];
    __shared__ float stage[8][16 * 52];

    int t = threadIdx.x;
    for (int i = t; i < 64 * 48; i += 256) {
        int r = i / 48, c = i % 48;
        wls[i] = (c < 40) ? Wl[r * 40 + c] : 0.0f;
    }
    if (t < 48) bs[t] = (t < 40) ? bl[t] : 0.0f;
    __syncthreads();

    int wave = t >> 5, lane = t & 31;
    int ntiles = (n + 15) >> 4;
    int tile = blockIdx.x * 8 + wave;
    if (tile >= ntiles) tile = ntiles - 1;
    int row0 = tile * 16;

    int am = lane & 15, ak = (lane >> 4) * 2;
    int bn = lane & 15, bk = (lane >> 4) * 2;

    v2f a[16];
    const float* ain = xemb + (long long)(row0 + am) * 64;
#pragma unroll
    for (int k = 0; k < 16; ++k) a[k] = *(const v2f*)(ain + k * 4 + ak);

    float* st = &stage[wave][0];
    int rb = (lane < 16) ? 0 : 8;
#pragma unroll
    for (int nt = 0; nt < 3; ++nt) {
        v8f c = {};
#pragma unroll
        for (int k = 0; k < 16; ++k) {
            int kr = k * 4 + bk;
            v2f b;
            b.x = wls[kr * 48 + nt * 16 + bn];
            b.y = wls[(kr + 1) * 48 + nt * 16 + bn];
            c = __builtin_amdgcn_wmma_f32_16x16x4_f32(false, a[k], false, b,
                                                      (short)0, c, false, false);
        }
        float bias = bs[nt * 16 + bn];
        int colw = nt * 16 + bn;
#pragma unroll
        for (int j = 0; j < 8; ++j) st[(rb + j) * 52 + colw] = c[j] + bias;
    }
    __syncthreads();

    if (lane < 16) {   // one row per lane: 40-wide log-softmax
        float m = -INFINITY;
        for (int c2 = 0; c2 < 40; ++c2) m = fmaxf(m, st[lane * 52 + c2]);
        float ssum = 0.0f;
        for (int c2 = 0; c2 < 40; ++c2) ssum += __expf(st[lane * 52 + c2] - m);
        float lse = m + __logf(ssum);
        long long grow = (long long)(row0 + lane) * 40;
        for (int c2 = 0; c2 < 40; ++c2) out[grow + c2] = st[lane * 52 + c2] - lse;
    }
}

// ---------------------------------------------------------------- launcher

extern "C" void kernel_launch(void* const* d_in, const int* in_sizes, int n_in,
                              void* d_out, int out_size, void* d_ws, size_t ws_size,
                              hipStream_t stream) {
    const float* x   = (const float*)d_in[0];
    const int*   ei  = (const int*)d_in[1];
    const float* W1a = (const float*)d_in[2];
    const float* b1a = (const float*)d_in[3];
    const float* W1b = (const float*)d_in[4];
    const float* b1b = (const float*)d_in[5];
    const float* Wg2 = (const float*)d_in[6];
    const float* bg2 = (const float*)d_in[7];
    const float* Wg3 = (const float*)d_in[8];
    const float* bg3 = (const float*)d_in[9];
    const float* Wl  = (const float*)d_in[10];
    const float* bl  = (const float*)d_in[11];

    int N = in_sizes[0] / 64;
    int E = in_sizes[1] / 2;
    const int* src = ei;
    const int* dst = ei + E;

    // workspace layout: [deg/dinv: N][bufA: N*64][bufB: N*64]
    float* deg = (float*)d_ws;
    size_t off = (((size_t)N * 4 + 255) & ~(size_t)255);
    float* bufA = (float*)((char*)d_ws + off);
    float* bufB = bufA + (size_t)N * 64;

    float* outLS = (float*)d_out;              // [N,40] log_softmax
    float* xemb  = outLS + (size_t)N * 40;     // [N,64] embedding output

    long long n64 = (long long)N * 64;
    int zb1 = (int)((N + 255) / 256);
    int zb64 = (int)((n64 + 255) / 256);
    long long ethreads = (long long)E * 32;
    int eblocks = (int)((ethreads + 255) / 256);
    int gblocks = (int)(((N + 15) / 16 + 7) / 8);

    // 1) GIN aggregation + degree
    zero_kernel<<<zb1, 256, 0, stream>>>(deg, N);
    zero_kernel<<<zb64, 256, 0, stream>>>(bufA, n64);
    gin_edge_kernel<<<eblocks, 256, 0, stream>>>(x, src, dst, bufA, deg, E);
    dinv_kernel<<<zb1, 256, 0, stream>>>(deg, N);

    // 2) GIN MLP (fused two 64x64 layers, relu on both): h1 -> bufB
    gemm64_wmma_kernel<<<gblocks, 256, 0, stream>>>(x, bufA, W1a, b1a, W1b, b1b,
                                                    bufB, N, 1, 1);

    // 3) GCN conv 2: hw2 = h1 @ Wg2 -> bufA ; scatter-normalized -> bufB
    gemm64_wmma_kernel<<<gblocks, 256, 0, stream>>>(bufB, nullptr, Wg2, nullptr,
                                                    nullptr, nullptr, bufA, N, 0, 0);
    gcn_init_kernel<<<zb64, 256, 0, stream>>>(bufA, deg, bg2, bufB, n64);
    gcn_edge_kernel<<<eblocks, 256, 0, stream>>>(bufA, src, dst, deg, bufB, E);

    // 4) GCN conv 3: hw3 = g2 @ Wg3 -> bufA ; result straight into x_emb slice
    gemm64_wmma_kernel<<<gblocks, 256, 0, stream>>>(bufB, nullptr, Wg3, nullptr,
                                                    nullptr, nullptr, bufA, N, 0, 0);
    gcn_init_kernel<<<zb64, 256, 0, stream>>>(bufA, deg, bg3, xemb, n64);
    gcn_edge_kernel<<<eblocks, 256, 0, stream>>>(bufA, src, dst, deg, xemb, E);

    // 5) classifier head + log_softmax
    final_kernel<<<gblocks, 256, 0, stream>>>(xemb, Wl, bl, outLS, N);
}